// SinkhornLoss_63187558858996
// MI455X (gfx1250) — compile-verified
//
#include <hip/hip_runtime.h>
#include <cstddef>

#define B_   8
#define NPTS 2048
#define DDIM 64

typedef __attribute__((ext_vector_type(2))) float v2f;
typedef __attribute__((ext_vector_type(8))) float v8f;

// ---------------- prep: squared norms and log-weights ----------------
__global__ void sq_norm_kernel(const float* __restrict__ X, float* __restrict__ out, int n) {
  int i = blockIdx.x * blockDim.x + threadIdx.x;
  if (i >= n) return;
  const float4* p = (const float4*)(X + (size_t)i * DDIM);
  float acc = 0.f;
#pragma unroll
  for (int t = 0; t < DDIM / 4; ++t) {
    float4 v = p[t];
    acc += v.x * v.x + v.y * v.y + v.z * v.z + v.w * v.w;
  }
  out[i] = acc;
}

__global__ void log_kernel(const float* __restrict__ w, float* __restrict__ out, int n) {
  int i = blockIdx.x * blockDim.x + threadIdx.x;
  if (i < n) out[i] = logf(w[i]);
}

// ---------------- cost matrix via fp32 WMMA ----------------
// C[b,i,j] = max(0, 0.5*(|x_i|^2 + |y_j|^2) - x_i . y_j)
// one 16x16 tile per wave, K=64 via 16 x V_WMMA_F32_16X16X4_F32
__global__ __launch_bounds__(256)
void cost_wmma_kernel(const float* __restrict__ X, const float* __restrict__ Y,
                      const float* __restrict__ nx, const float* __restrict__ ny,
                      float* __restrict__ C, float* __restrict__ CT) {
  const int lane = threadIdx.x & 31;
  const int wave = threadIdx.x >> 5;
  const int b  = blockIdx.z;
  const int tj = blockIdx.x;                 // column tile
  const int ti = blockIdx.y * 8 + wave;      // row tile
  const int r0 = ti * 16, c0 = tj * 16;
  const int mr = lane & 15;                  // row (A) / col (B) within tile
  const int kh = (lane >> 4) << 1;           // K sub-offset: 0 or 2

  const float* xa = X + ((size_t)b * NPTS + r0 + mr) * DDIM + kh;
  const float* yb = Y + ((size_t)b * NPTS + c0 + mr) * DDIM + kh;

  v2f a[16], bm[16];
#pragma unroll
  for (int k = 0; k < 16; ++k) {
    float2 av = *(const float2*)(xa + k * 4);
    float2 bv = *(const float2*)(yb + k * 4);
    a[k]  = (v2f){av.x, av.y};
    bm[k] = (v2f){bv.x, bv.y};
  }

  v8f acc = {};
#pragma unroll
  for (int k = 0; k < 16; ++k)
    acc = __builtin_amdgcn_wmma_f32_16x16x4_f32(false, a[k], false, bm[k],
                                                (short)0, acc, false, false);

  // D layout: VGPR r -> M = r (lanes 0-15) or r+8 (lanes 16-31); N = lane&15
  const int col   = c0 + (lane & 15);
  const int rbase = r0 + ((lane >> 4) << 3);
  const float nyc = ny[(size_t)b * NPTS + col];
#pragma unroll
  for (int r = 0; r < 8; ++r) {
    int row = rbase + r;
    float v = fmaxf(0.5f * (nx[(size_t)b * NPTS + row] + nyc) - acc[r], 0.f);
    C[((size_t)b * NPTS + row) * NPTS + col] = v;
    if (CT) CT[((size_t)b * NPTS + col) * NPTS + row] = v;
  }
}

// ---------------- row-wise softmin: out_i = -eps * lse_j(h_j - C_ij/eps) ----------------
// one wave per row; base-2 online logsumexp; optional 0.5*(old+new) averaging
__global__ __launch_bounds__(256)
void softmin_rows_kernel(const float* __restrict__ C,
                         const float* __restrict__ hbase,
                         const float* __restrict__ hextra,   // nullable: h += hextra/eps
                         const float* __restrict__ oldf,     // nullable when do_avg==0
                         float* __restrict__ outf,
                         float eps, float inv_eps, int do_avg) {
  const int lane = threadIdx.x & 31;
  const int row  = blockIdx.x * 8 + (threadIdx.x >> 5);
  const int b    = row >> 11;                 // NPTS == 2048
  const float4* Cr = (const float4*)(C + (size_t)row * NPTS);
  const float4* Hb = (const float4*)(hbase + ((size_t)b << 11));
  const float4* He = hextra ? (const float4*)(hextra + ((size_t)b << 11)) : nullptr;

  const float LOG2E = 1.4426950408889634f;
  const float cs = inv_eps * LOG2E;
  float m = -3.0e38f, s = 0.f;

#pragma unroll 4
  for (int t = 0; t < NPTS / (4 * 32); ++t) {
    int idx = lane + (t << 5);
    float4 cv = Cr[idx];
    float4 hv = Hb[idx];
    float h0 = hv.x, h1 = hv.y, h2 = hv.z, h3 = hv.w;
    if (He) {
      float4 ev = He[idx];
      h0 = fmaf(ev.x, inv_eps, h0); h1 = fmaf(ev.y, inv_eps, h1);
      h2 = fmaf(ev.z, inv_eps, h2); h3 = fmaf(ev.w, inv_eps, h3);
    }
    float v0 = h0 * LOG2E - cv.x * cs;
    float v1 = h1 * LOG2E - cv.y * cs;
    float v2 = h2 * LOG2E - cv.z * cs;
    float v3 = h3 * LOG2E - cv.w * cs;
    float m4 = fmaxf(fmaxf(v0, v1), fmaxf(v2, v3));
    float nm = fmaxf(m, m4);
    s = s * exp2f(m - nm)
      + exp2f(v0 - nm) + exp2f(v1 - nm) + exp2f(v2 - nm) + exp2f(v3 - nm);
    m = nm;
  }

  // wave32 cross-lane (max, scaled-sum) reduce
#pragma unroll
  for (int off = 16; off > 0; off >>= 1) {
    float om = __shfl_xor(m, off, 32);
    float os = __shfl_xor(s, off, 32);
    float nm = fmaxf(m, om);
    s = s * exp2f(m - nm) + os * exp2f(om - nm);
    m = nm;
  }

  if (lane == 0) {
    const float LN2 = 0.6931471805599453f;
    float f = -eps * (m + log2f(s)) * LN2;
    outf[row] = do_avg ? 0.5f * (oldf[row] + f) : f;
  }
}

// ---------------- final weighted reduction ----------------
__global__ void loss_reduce_kernel(const float* __restrict__ w1, const float* __restrict__ w2,
                                   const float* __restrict__ fab, const float* __restrict__ faa,
                                   const float* __restrict__ gba, const float* __restrict__ gbb,
                                   float* __restrict__ out) {
  __shared__ float red[256];
  float acc = 0.f;
  const int n = B_ * NPTS;
  for (int i = threadIdx.x; i < n; i += 256)
    acc += w1[i] * (fab[i] - faa[i]) + w2[i] * (gba[i] - gbb[i]);
  red[threadIdx.x] = acc;
  __syncthreads();
  for (int off = 128; off > 0; off >>= 1) {
    if (threadIdx.x < off) red[threadIdx.x] += red[threadIdx.x + off];
    __syncthreads();
  }
  if (threadIdx.x == 0) out[0] = red[0] / (float)B_;
}

extern "C" void kernel_launch(void* const* d_in, const int* in_sizes, int n_in,
                              void* d_out, int out_size, void* d_ws, size_t ws_size,
                              hipStream_t stream) {
  (void)in_sizes; (void)n_in; (void)out_size; (void)ws_size;
  const float* X  = (const float*)d_in[0];
  const float* Y  = (const float*)d_in[1];
  const float* W1 = (const float*)d_in[2];
  const float* W2 = (const float*)d_in[3];
  float* out = (float*)d_out;

  const size_t MAT = (size_t)B_ * NPTS * NPTS;
  const size_t VEC = (size_t)B_ * NPTS;
  float* ws = (float*)d_ws;
  float* C_xy = ws;
  float* C_yx = ws + MAT;
  float* C_xx = ws + 2 * MAT;
  float* C_yy = ws + 3 * MAT;
  float* sm = ws + 4 * MAT;
  float* sqnx = sm;  sm += VEC;
  float* sqny = sm;  sm += VEC;
  float* a_log = sm; sm += VEC;
  float* b_log = sm; sm += VEC;
  float* cur[4]; float* nxt[4];
  for (int i = 0; i < 4; ++i) { cur[i] = sm; sm += VEC; }
  for (int i = 0; i < 4; ++i) { nxt[i] = sm; sm += VEC; }

  dim3 blk(256);
  sq_norm_kernel<<<dim3((unsigned)(VEC / 256)), blk, 0, stream>>>(X, sqnx, (int)VEC);
  sq_norm_kernel<<<dim3((unsigned)(VEC / 256)), blk, 0, stream>>>(Y, sqny, (int)VEC);
  log_kernel<<<dim3((unsigned)(VEC / 256)), blk, 0, stream>>>(W1, a_log, (int)VEC);
  log_kernel<<<dim3((unsigned)(VEC / 256)), blk, 0, stream>>>(W2, b_log, (int)VEC);

  dim3 ggrid(NPTS / 16, NPTS / 16 / 8, B_);
  cost_wmma_kernel<<<ggrid, blk, 0, stream>>>(X, Y, sqnx, sqny, C_xy, C_yx);
  cost_wmma_kernel<<<ggrid, blk, 0, stream>>>(X, X, sqnx, sqnx, C_xx, nullptr);
  cost_wmma_kernel<<<ggrid, blk, 0, stream>>>(Y, Y, sqny, sqny, C_yy, nullptr);

  // eps schedule: diameter^2=4096 down by scaling^2=0.25 to blur^2=0.0025
  float eps_list[24]; int ne = 0;
  { double e = 4096.0; const double target = 0.05 * 0.05;
    while (e > target) { eps_list[ne++] = (float)e; e *= 0.25; }
    eps_list[ne++] = (float)target; }

  const unsigned sblocks = (unsigned)(VEC / 8);   // 8 waves (rows) per block
  auto softmin = [&](const float* C, const float* hb, const float* he,
                     const float* oldf, float* outf, float eps, int avg) {
    softmin_rows_kernel<<<dim3(sblocks), blk, 0, stream>>>(
        C, hb, he, oldf, outf, eps, 1.0f / eps, avg);
  };

  // init at eps0:  f_ab=cur[0], g_ba=cur[1], f_aa=cur[2], g_bb=cur[3]
  float e0 = eps_list[0];
  softmin(C_xy, b_log, nullptr, nullptr, cur[0], e0, 0);
  softmin(C_yx, a_log, nullptr, nullptr, cur[1], e0, 0);
  softmin(C_xx, a_log, nullptr, nullptr, cur[2], e0, 0);
  softmin(C_yy, b_log, nullptr, nullptr, cur[3], e0, 0);

  // scan over full eps list with 0.5*(old+new) averaging
  for (int k = 0; k < ne; ++k) {
    float e = eps_list[k];
    softmin(C_xy, b_log, cur[1], cur[0], nxt[0], e, 1);
    softmin(C_yx, a_log, cur[0], cur[1], nxt[1], e, 1);
    softmin(C_xx, a_log, cur[2], cur[2], nxt[2], e, 1);
    softmin(C_yy, b_log, cur[3], cur[3], nxt[3], e, 1);
    for (int i = 0; i < 4; ++i) { float* t = cur[i]; cur[i] = nxt[i]; nxt[i] = t; }
  }

  // final extrapolation at eps = blur^2 (no averaging)
  float ef = eps_list[ne - 1];
  softmin(C_xy, b_log, cur[1], nullptr, nxt[0], ef, 0);
  softmin(C_yx, a_log, cur[0], nullptr, nxt[1], ef, 0);
  softmin(C_xx, a_log, cur[2], nullptr, nxt[2], ef, 0);
  softmin(C_yy, b_log, cur[3], nullptr, nxt[3], ef, 0);

  loss_reduce_kernel<<<dim3(1), blk, 0, stream>>>(W1, W2, nxt[0], nxt[2], nxt[1], nxt[3], out);
}